// SAGE_23794118820012
// MI455X (gfx1250) — compile-verified
//
#include <hip/hip_runtime.h>
#include <math.h>

#define NN 100000
#define NE 600000
#define DIN 256
#define DH 128
#define DC 47
#define ALPHA 0.1f

typedef __attribute__((ext_vector_type(2))) float v2f;
typedef __attribute__((ext_vector_type(8))) float v8f;

__device__ __forceinline__ v8f wmma4(v2f a, v2f b, v8f c) {
    // D = A(16x4) * B(4x16) + C, full f32
    return __builtin_amdgcn_wmma_f32_16x16x4_f32(false, a, false, b, (short)0, c,
                                                 false, false);
}

// ---------- utility kernels ----------
__global__ void zero_f4_kernel(float4* __restrict__ p, int n4) {
    int i = blockIdx.x * blockDim.x + threadIdx.x;
    if (i < n4) p[i] = make_float4(0.f, 0.f, 0.f, 0.f);
}

__global__ void degree_kernel(const int* __restrict__ src, const int* __restrict__ dst,
                              float* __restrict__ deg_out, float* __restrict__ deg_in) {
    int i = blockIdx.x * blockDim.x + threadIdx.x;
    if (i < NE) {
        unsafeAtomicAdd(&deg_out[src[i]], 1.0f);
        unsafeAtomicAdd(&deg_in[dst[i]], 1.0f);
    }
}

__global__ void finalize_norm_kernel(float* __restrict__ v, int n) {
    int i = blockIdx.x * blockDim.x + threadIdx.x;
    if (i < n) v[i] = rsqrtf(fmaxf(v[i], 1.0f));
}

// ---------- x0 = x @ lin1_w^T   [NN,256]x[256,128] ----------
__global__ __launch_bounds__(256) void gemm_x0_kernel(const float* __restrict__ x,
                                                      const float* __restrict__ w,
                                                      float* __restrict__ x0) {
    __shared__ float xs[16 * DIN];  // 16 KB
    const int node0 = blockIdx.x * 16;
    const int t = threadIdx.x;
    for (int i = t; i < 16 * DIN / 4; i += 256) {
        int row = i >> 6;            // DIN/4 = 64 float4 per row
        int c4 = i & 63;
        ((float4*)xs)[i] = *((const float4*)(x + (size_t)(node0 + row) * DIN) + c4);
    }
    __syncthreads();
    const int wave = t >> 5, lane = t & 31;
    const int half = lane >> 4, lm = lane & 15, kb = half * 2;
    const int colg = wave * 16 + lm;  // 8 waves -> 128 output cols
    v8f acc = {0.f, 0.f, 0.f, 0.f, 0.f, 0.f, 0.f, 0.f};
    for (int k0 = 0; k0 < DIN; k0 += 4) {
        v2f a, b;
        const float* ap = &xs[lm * DIN + k0 + kb];
        a.x = ap[0]; a.y = ap[1];
        // B[k][n] = lin1_w[n][k]  (transposed weight access)
        b.x = w[colg * DIN + k0 + kb];
        b.y = w[colg * DIN + k0 + kb + 1];
        acc = wmma4(a, b, acc);
    }
    for (int r = 0; r < 8; ++r) {
        int m = r + half * 8;
        x0[(size_t)(node0 + m) * DH + colg] = acc[r];
    }
}

// ---------- edge scatter: agg[dst] += h[src] * nsrc[src] ----------
__global__ __launch_bounds__(256) void scatter_kernel(const float* __restrict__ h,
                                                      const int* __restrict__ src,
                                                      const int* __restrict__ dst,
                                                      const float* __restrict__ nsrc,
                                                      float* __restrict__ agg) {
    int tid = blockIdx.x * 256 + threadIdx.x;
    int e = tid >> 5;          // 32 threads per edge (128 feats / 4)
    int c4 = tid & 31;
    if (e < NE) {
        int s = src[e], d = dst[e];
        float sc = nsrc[s];
        float4 v = *((const float4*)(h + (size_t)s * DH) + c4);
        float* base = agg + (size_t)d * DH + c4 * 4;
        unsafeAtomicAdd(base + 0, v.x * sc);
        unsafeAtomicAdd(base + 1, v.y * sc);
        unsafeAtomicAdd(base + 2, v.z * sc);
        unsafeAtomicAdd(base + 3, v.w * sc);
    }
}

// ---------- fused GCN2 layer ----------
// h_out = (1-b)*(feat+f0) + b*(feat@W1 + f0@W2) + bias
// feat = agg*ndst*(1-alpha), f0 = x0*alpha  (folded into LDS staging)
__global__ __launch_bounds__(256) void layer_kernel(const float* __restrict__ agg,
                                                    const float* __restrict__ x0,
                                                    const float* __restrict__ w1,
                                                    const float* __restrict__ w2,
                                                    const float* __restrict__ bias,
                                                    const float* __restrict__ ndst,
                                                    float beta, float* __restrict__ h) {
    __shared__ float feat_s[16 * DH];  // 8 KB
    __shared__ float f0_s[16 * DH];    // 8 KB
    const int node0 = blockIdx.x * 16;
    const int t = threadIdx.x;
    for (int i = t; i < 16 * DH / 4; i += 256) {
        int row = i >> 5;  // 32 float4 per row
        int c4 = i & 31;
        int node = node0 + row;
        float s = ndst[node] * (1.0f - ALPHA);
        float4 av = *((const float4*)(agg + (size_t)node * DH) + c4);
        float4 xv = *((const float4*)(x0 + (size_t)node * DH) + c4);
        ((float4*)feat_s)[i] = make_float4(av.x * s, av.y * s, av.z * s, av.w * s);
        ((float4*)f0_s)[i] = make_float4(xv.x * ALPHA, xv.y * ALPHA, xv.z * ALPHA, xv.w * ALPHA);
    }
    __syncthreads();
    const int wave = t >> 5, lane = t & 31;
    const int half = lane >> 4, lm = lane & 15, kb = half * 2;
    const int colg = wave * 16 + lm;
    const float bv = bias[colg];
    v8f acc1 = {0.f, 0.f, 0.f, 0.f, 0.f, 0.f, 0.f, 0.f};
    v8f acc2 = {0.f, 0.f, 0.f, 0.f, 0.f, 0.f, 0.f, 0.f};
    for (int k0 = 0; k0 < DH; k0 += 4) {
        v2f a1, a2, b1, b2;
        const float* p1 = &feat_s[lm * DH + k0 + kb];
        a1.x = p1[0]; a1.y = p1[1];
        const float* p2 = &f0_s[lm * DH + k0 + kb];
        a2.x = p2[0]; a2.y = p2[1];
        b1.x = w1[(k0 + kb) * DH + colg];       // row-major W1: B[k][n]
        b1.y = w1[(k0 + kb + 1) * DH + colg];
        b2.x = w2[(k0 + kb) * DH + colg];
        b2.y = w2[(k0 + kb + 1) * DH + colg];
        acc1 = wmma4(a1, b1, acc1);
        acc2 = wmma4(a2, b2, acc2);
    }
    for (int r = 0; r < 8; ++r) {
        int m = r + half * 8;
        float lin = feat_s[m * DH + colg] + f0_s[m * DH + colg];
        h[(size_t)(node0 + m) * DH + colg] =
            (1.0f - beta) * lin + beta * (acc1[r] + acc2[r]) + bv;
    }
}

// ---------- out = h @ lin2_w^T   [NN,128]x[128,47] ----------
__global__ __launch_bounds__(96) void out_kernel(const float* __restrict__ h,
                                                 const float* __restrict__ w,
                                                 float* __restrict__ out) {
    __shared__ float hs[16 * DH];  // 8 KB
    const int node0 = blockIdx.x * 16;
    const int t = threadIdx.x;
    for (int i = t; i < 16 * DH / 4; i += 96) {
        int row = i >> 5;
        int c4 = i & 31;
        ((float4*)hs)[i] = *((const float4*)(h + (size_t)(node0 + row) * DH) + c4);
    }
    __syncthreads();
    const int wave = t >> 5, lane = t & 31;
    const int half = lane >> 4, lm = lane & 15, kb = half * 2;
    const int colg = wave * 16 + lm;   // 3 waves -> 48 cols, guard at 47
    const bool valid = (colg < DC);
    v8f acc = {0.f, 0.f, 0.f, 0.f, 0.f, 0.f, 0.f, 0.f};
    for (int k0 = 0; k0 < DH; k0 += 4) {
        v2f a, b;
        const float* p = &hs[lm * DH + k0 + kb];
        a.x = p[0]; a.y = p[1];
        b.x = valid ? w[colg * DH + k0 + kb] : 0.0f;       // B[k][n] = lin2_w[n][k]
        b.y = valid ? w[colg * DH + k0 + kb + 1] : 0.0f;
        acc = wmma4(a, b, acc);  // EXEC stays all-ones; masking via operand select
    }
    if (valid) {
        for (int r = 0; r < 8; ++r) {
            int m = r + half * 8;
            out[(size_t)(node0 + m) * DC + colg] = acc[r];
        }
    }
}

extern "C" void kernel_launch(void* const* d_in, const int* in_sizes, int n_in,
                              void* d_out, int out_size, void* d_ws, size_t ws_size,
                              hipStream_t stream) {
    const float* x = (const float*)d_in[0];
    const int* src = (const int*)d_in[1];
    const int* dst = (const int*)d_in[2];
    const float* lin1_w = (const float*)d_in[3];
    const float* lin2_w = (const float*)d_in[4];
    const float* w1[3] = {(const float*)d_in[5], (const float*)d_in[8], (const float*)d_in[11]};
    const float* w2[3] = {(const float*)d_in[6], (const float*)d_in[9], (const float*)d_in[12]};
    const float* bs[3] = {(const float*)d_in[7], (const float*)d_in[10], (const float*)d_in[13]};

    float* ws = (float*)d_ws;
    const size_t NH = (size_t)NN * DH;   // 12.8M floats
    float* x0 = ws;                      // [NN, DH]
    float* h = ws + NH;                  // [NN, DH]
    float* agg = ws + 2 * NH;            // [NN, DH]
    float* nsrc = ws + 3 * NH;           // [NN]
    float* ndst = nsrc + NN;             // [NN], contiguous with nsrc

    const int MT = NN / 16;  // 6250 node tiles (exact)

    // degrees -> deg^-1/2 (nsrc/ndst buffers hold degree, finalized in place)
    zero_f4_kernel<<<(2 * NN / 4 + 255) / 256, 256, 0, stream>>>((float4*)nsrc, 2 * NN / 4);
    degree_kernel<<<(NE + 255) / 256, 256, 0, stream>>>(src, dst, nsrc, ndst);
    finalize_norm_kernel<<<(2 * NN + 255) / 256, 256, 0, stream>>>(nsrc, 2 * NN);

    // x0 = x @ lin1_w^T
    gemm_x0_kernel<<<MT, 256, 0, stream>>>(x, lin1_w, x0);

    const float betas[3] = {logf(2.0f), logf(1.5f), logf(4.0f / 3.0f)};
    const float* hcur = x0;  // h == x0 for layer 1
    for (int l = 0; l < 3; ++l) {
        zero_f4_kernel<<<(int)(NH / 4 / 256), 256, 0, stream>>>((float4*)agg, (int)(NH / 4));
        scatter_kernel<<<(int)(NE * 32 / 256), 256, 0, stream>>>(hcur, src, dst, nsrc, agg);
        layer_kernel<<<MT, 256, 0, stream>>>(agg, x0, w1[l], w2[l], bs[l], ndst, betas[l], h);
        hcur = h;
    }

    out_kernel<<<MT, 96, 0, stream>>>(h, lin2_w, (float*)d_out);
}